// LocalTrittention_39101382263023
// MI455X (gfx1250) — compile-verified
//
#include <hip/hip_runtime.h>
#include <hip/hip_bf16.h>

typedef __attribute__((ext_vector_type(2))) float v2f;
typedef __attribute__((ext_vector_type(8))) float v8f;

#define D_MODEL 768
#define N_HEADS 12
#define D_HEAD  64
#define WINDOW  32
#define SEQ     1024
#define BS      2
#define NW      (SEQ / WINDOW)          // 32 windows
#define BH      (BS * N_HEADS)          // 24 batch-heads
#define NCOLS   (5 * N_HEADS * D_HEAD)  // 3840
#define IGNORE_VAL (-1000000.0f)

// Async global->LDS copy of 16 bytes per lane (CDNA5 GLOBAL_LOAD_ASYNC_TO_LDS_B128,
// tracked by ASYNCcnt; the DMA writes LDS directly, no VGPR round trip).
// lds_off: wave-relative LDS byte address (low 32 bits of a generic LDS pointer,
// per the aperture mapping). Must be 16B aligned.
__device__ __forceinline__ void async_copy_b128(const void* gsrc, void* lds_dst) {
    unsigned dst = (unsigned)(unsigned long long)lds_dst;
    asm volatile("global_load_async_to_lds_b128 %0, %1, off"
                 :: "v"(dst), "v"(gsrc) : "memory");
}
__device__ __forceinline__ void wait_async_le8() {
    asm volatile("s_wait_asynccnt 0x8" ::: "memory");
}
__device__ __forceinline__ void wait_async_0() {
    asm volatile("s_wait_asynccnt 0x0" ::: "memory");
}

// ---------------------------------------------------------------------------
// Generic fp32 WMMA GEMM: C = A(MxK) * B(KxN) + bias(N)
// Block tile 64x64, 128 threads (4 wave32), wave w owns rows [16w,16w+16).
// K staged in 32-wide chunks through double-buffered LDS filled by async
// global->LDS DMA; copy of chunk i+1 overlaps WMMA on chunk i.
// MODE 0: scatter epilogue into abcde layout (5, BH, t, d) for attention.
// MODE 1: plain row-major output.
// ---------------------------------------------------------------------------
template <int MODE>
__global__ __launch_bounds__(128)
void gemm_wmma_f32(const float* __restrict__ A, const float* __restrict__ B,
                   const float* __restrict__ bias, float* __restrict__ out,
                   int M, int N, int K) {
    __shared__ __align__(16) float As[2][64 * 36];  // 64 rows x 32 k (144B row pitch)
    __shared__ __align__(16) float Bs[2][32 * 68];  // 32 k x 64 n  (272B row pitch)

    const int tid  = threadIdx.x;
    const int wave = tid >> 5;
    const int lane = tid & 31;
    const int half = lane >> 4;
    const int lcol = lane & 15;
    const int bm = blockIdx.x * 64;
    const int bn = blockIdx.y * 64;

    // Issue the 8 async B128 ops (per wave) that fill one LDS buffer from chunk k0.
    auto issue_chunk = [&](int buf, int k0) {
        #pragma unroll
        for (int it = 0; it < 4; ++it) {            // A: 64x32 f = 512 float4
            int idx = tid + it * 128;
            int r = idx >> 3, c4 = (idx & 7) * 4;   // row 0..63, col 0..28
            async_copy_b128(&A[(size_t)(bm + r) * K + k0 + c4], &As[buf][r * 36 + c4]);
        }
        #pragma unroll
        for (int it = 0; it < 4; ++it) {            // B: 32x64 f = 512 float4
            int idx = tid + it * 128;
            int r = idx >> 4, c4 = (idx & 15) * 4;  // row 0..31, col 0..60
            async_copy_b128(&B[(size_t)(k0 + r) * N + bn + c4], &Bs[buf][r * 68 + c4]);
        }
    };

    v8f acc[4] = {};
    const int nch = K / 32;

    issue_chunk(0, 0);
    for (int ci = 0; ci < nch; ++ci) {
        if (ci + 1 < nch) {
            // Overwrites the buffer read in the previous iteration; the trailing
            // __syncthreads of that compute phase (s_wait_dscnt 0 + barrier)
            // guarantees every wave finished reading it.
            issue_chunk((ci + 1) & 1, (ci + 1) * 32);
            wait_async_le8();    // oldest 8 (= current buffer) complete
        } else {
            wait_async_0();
        }
        __syncthreads();

        const float* as = As[ci & 1];
        const float* bs = Bs[ci & 1];
        #pragma unroll
        for (int ks = 0; ks < 8; ++ks) {
            const int d0 = ks * 4 + 2 * half;        // ISA A/B 16x4 fragment layout
            v2f a;
            a.x = as[(wave * 16 + lcol) * 36 + d0];
            a.y = as[(wave * 16 + lcol) * 36 + d0 + 1];
            #pragma unroll
            for (int nt = 0; nt < 4; ++nt) {
                v2f b;
                b.x = bs[d0 * 68 + nt * 16 + lcol];
                b.y = bs[(d0 + 1) * 68 + nt * 16 + lcol];
                acc[nt] = __builtin_amdgcn_wmma_f32_16x16x4_f32(
                    false, a, false, b, (short)0, acc[nt], false, false);
            }
        }
        __syncthreads();
    }

    // Epilogue. C/D layout: element v -> M = v + 8*half, N = lane&15.
    #pragma unroll
    for (int nt = 0; nt < 4; ++nt) {
        #pragma unroll
        for (int v = 0; v < 8; ++v) {
            int m = bm + wave * 16 + v + 8 * half;
            int n = bn + nt * 16 + lcol;
            float val = acc[nt][v] + bias[n];
            if (MODE == 0) {
                // m -> (batch, t); n -> (s, head, d); dst layout (5, BH, SEQ, D_HEAD)
                int b_idx = m >> 10, t = m & 1023;
                int s = n / D_MODEL, rem = n - s * D_MODEL;
                int h = rem >> 6, d = rem & 63;
                out[(((size_t)(s * BH + b_idx * N_HEADS + h) * SEQ + t) << 6) + d] = val;
            } else {
                out[(size_t)m * N + n] = val;
            }
        }
    }
}

// ---------------------------------------------------------------------------
// Scores kernel: one block per (window n, batch-head x).
// attn_i(64x64) = H_i @ la_b^T with H_i[j,d] = c[i,d]*la_a[j,d], via fp32 WMMA.
// Mask + softmax in registers (wave32 shfl_xor reductions), write sj/sk
// marginals (128 floats per query) to scratch.
// ---------------------------------------------------------------------------
__global__ __launch_bounds__(256)
void tritt_scores(const float* __restrict__ wsA, float* __restrict__ wsM) {
    const int n = blockIdx.x;   // window
    const int x = blockIdx.y;   // batch-head

    __shared__ float sc[32 * 65];   // c   (32 x 64)
    __shared__ float sa[64 * 65];   // la_a(64 x 64)
    __shared__ float sb[64 * 65];   // la_b(64 x 64)

    const float* baseA = wsA + (size_t)(0 * BH + x) * SEQ * D_HEAD;
    const float* baseB = wsA + (size_t)(1 * BH + x) * SEQ * D_HEAD;
    const float* baseC = wsA + (size_t)(2 * BH + x) * SEQ * D_HEAD;

    const int tid = threadIdx.x;
    for (int idx = tid; idx < 32 * 64; idx += 256) {
        int i = idx >> 6, d = idx & 63;
        sc[i * 65 + d] = baseC[(size_t)(n * 32 + i) * 64 + d];
    }
    for (int idx = tid; idx < 64 * 64; idx += 256) {
        int j = idx >> 6, d = idx & 63;
        int t = n * 32 - 32 + j;                    // look_around(back=1)
        sa[j * 65 + d] = (t < 0) ? 0.0f : baseA[(size_t)t * 64 + d];
        sb[j * 65 + d] = (t < 0) ? 0.0f : baseB[(size_t)t * 64 + d];
    }
    __syncthreads();

    const int wave = tid >> 5, lane = tid & 31;
    const int half = lane >> 4, lcol = lane & 15;

    for (int q = 0; q < 4; ++q) {
        const int i  = wave * 4 + q;
        const int qi = n * 32 + i;

        v8f acc[4][4] = {};
        #pragma unroll
        for (int ks = 0; ks < 16; ++ks) {
            const int d0 = ks * 4 + 2 * half;
            const float c0 = sc[i * 65 + d0];
            const float c1 = sc[i * 65 + d0 + 1];
            v2f af[4], bf[4];
            #pragma unroll
            for (int jm = 0; jm < 4; ++jm) {
                af[jm].x = sa[(jm * 16 + lcol) * 65 + d0] * c0;
                af[jm].y = sa[(jm * 16 + lcol) * 65 + d0 + 1] * c1;
            }
            #pragma unroll
            for (int nt = 0; nt < 4; ++nt) {
                bf[nt].x = sb[(nt * 16 + lcol) * 65 + d0];
                bf[nt].y = sb[(nt * 16 + lcol) * 65 + d0 + 1];
            }
            #pragma unroll
            for (int jm = 0; jm < 4; ++jm)
                #pragma unroll
                for (int nt = 0; nt < 4; ++nt)
                    acc[jm][nt] = __builtin_amdgcn_wmma_f32_16x16x4_f32(
                        false, af[jm], false, bf[nt], (short)0, acc[jm][nt], false, false);
        }

        // Mask + scale. Element (jm,nt,v): j = jm*16+v+8*half, kk = nt*16+lcol.
        float mx = -3.4e38f;
        #pragma unroll
        for (int jm = 0; jm < 4; ++jm)
            #pragma unroll
            for (int nt = 0; nt < 4; ++nt)
                #pragma unroll
                for (int v = 0; v < 8; ++v) {
                    int j  = jm * 16 + v + 8 * half;
                    int kk = nt * 16 + lcol;
                    int tj = n * 32 - 32 + j;  int kj = (tj < 0) ? 0 : tj;
                    int tk = n * 32 - 32 + kk; int lk = (tk < 0) ? 0 : tk;
                    float raw = acc[jm][nt][v];
                    bool masked = (qi < lk) || (lk <= kj) || (raw == 0.0f);
                    float val = (masked ? IGNORE_VAL : raw) * (1.0f / (float)D_HEAD);
                    acc[jm][nt][v] = val;
                    mx = fmaxf(mx, val);
                }
        #pragma unroll
        for (int off = 16; off >= 1; off >>= 1)
            mx = fmaxf(mx, __shfl_xor(mx, off, 32));

        float sum = 0.0f;
        #pragma unroll
        for (int jm = 0; jm < 4; ++jm)
            #pragma unroll
            for (int nt = 0; nt < 4; ++nt)
                #pragma unroll
                for (int v = 0; v < 8; ++v) {
                    float p = __expf(acc[jm][nt][v] - mx);
                    acc[jm][nt][v] = p;
                    sum += p;
                }
        #pragma unroll
        for (int off = 16; off >= 1; off >>= 1)
            sum += __shfl_xor(sum, off, 32);
        const float inv = 1.0f / sum;

        // Marginals: sj[j] = sum_k p, sk[k] = sum_j p (normalize at the end).
        float pj[4][8];
        float pk[4] = {0.0f, 0.0f, 0.0f, 0.0f};
        #pragma unroll
        for (int jm = 0; jm < 4; ++jm)
            #pragma unroll
            for (int v = 0; v < 8; ++v) {
                float s = 0.0f;
                #pragma unroll
                for (int nt = 0; nt < 4; ++nt) s += acc[jm][nt][v];
                pj[jm][v] = s;
            }
        #pragma unroll
        for (int nt = 0; nt < 4; ++nt) {
            float s = 0.0f;
            #pragma unroll
            for (int jm = 0; jm < 4; ++jm)
                #pragma unroll
                for (int v = 0; v < 8; ++v) s += acc[jm][nt][v];
            pk[nt] = s;
        }
        // sj: reduce across the 16 lanes of each half (columns kk within half).
        #pragma unroll
        for (int jm = 0; jm < 4; ++jm)
            #pragma unroll
            for (int v = 0; v < 8; ++v) {
                float s = pj[jm][v];
                s += __shfl_xor(s, 1, 32);
                s += __shfl_xor(s, 2, 32);
                s += __shfl_xor(s, 4, 32);
                s += __shfl_xor(s, 8, 32);
                pj[jm][v] = s * inv;
            }
        // sk: combine the two halves (j halves).
        #pragma unroll
        for (int nt = 0; nt < 4; ++nt)
            pk[nt] = (pk[nt] + __shfl_xor(pk[nt], 16, 32)) * inv;

        float* mrow = wsM + ((size_t)((x * NW + n) * 32 + i)) * 128;
        if (lcol == 0) {
            #pragma unroll
            for (int jm = 0; jm < 4; ++jm)
                #pragma unroll
                for (int v = 0; v < 8; ++v)
                    mrow[jm * 16 + v + 8 * half] = pj[jm][v];
        }
        if (lane < 16) {
            #pragma unroll
            for (int nt = 0; nt < 4; ++nt)
                mrow[64 + nt * 16 + lane] = pk[nt];
        }
    }
}

// ---------------------------------------------------------------------------
// z kernel: z(32x64) = [sj|sk](32x128) @ [la_d; la_e](128x64) per window,
// one WMMA tile per wave, scattered into (bs, t, h*64+d) for the output GEMM.
// ---------------------------------------------------------------------------
__global__ __launch_bounds__(256)
void tritt_z(const float* __restrict__ wsA, const float* __restrict__ wsM,
             float* __restrict__ wsZ) {
    const int n = blockIdx.x;
    const int x = blockIdx.y;

    __shared__ float S[32 * 130];   // marginals (32 x 128)
    __shared__ float V[128 * 65];   // [la_d; la_e] (128 x 64)

    const float* baseD = wsA + (size_t)(3 * BH + x) * SEQ * D_HEAD;
    const float* baseE = wsA + (size_t)(4 * BH + x) * SEQ * D_HEAD;
    const float* mbase = wsM + (size_t)((x * NW + n) * 32) * 128;

    const int tid = threadIdx.x;
    for (int idx = tid; idx < 32 * 128; idx += 256) {
        int i = idx >> 7, c = idx & 127;
        S[i * 130 + c] = mbase[i * 128 + c];
    }
    for (int idx = tid; idx < 128 * 64; idx += 256) {
        int r = idx >> 6, d = idx & 63;
        int j = (r < 64) ? r : (r - 64);
        int t = n * 32 - 32 + j;
        const float* base = (r < 64) ? baseD : baseE;
        V[r * 65 + d] = (t < 0) ? 0.0f : base[(size_t)t * 64 + d];
    }
    __syncthreads();

    const int wave = tid >> 5, lane = tid & 31;
    const int half = lane >> 4, lcol = lane & 15;
    const int mt = wave >> 2, nt = wave & 3;   // 2 x 4 tiles, one per wave

    v8f acc = {};
    #pragma unroll
    for (int ks = 0; ks < 32; ++ks) {
        const int d0 = ks * 4 + 2 * half;
        v2f a, b;
        a.x = S[(mt * 16 + lcol) * 130 + d0];
        a.y = S[(mt * 16 + lcol) * 130 + d0 + 1];
        b.x = V[d0 * 65 + nt * 16 + lcol];
        b.y = V[(d0 + 1) * 65 + nt * 16 + lcol];
        acc = __builtin_amdgcn_wmma_f32_16x16x4_f32(
            false, a, false, b, (short)0, acc, false, false);
    }

    const int b_idx = x / N_HEADS;
    const int h = x - b_idx * N_HEADS;
    #pragma unroll
    for (int v = 0; v < 8; ++v) {
        int i = mt * 16 + v + 8 * half;
        int d = nt * 16 + lcol;
        int t = n * 32 + i;
        wsZ[((size_t)(b_idx * SEQ + t)) * D_MODEL + h * 64 + d] = acc[v];
    }
}

// ---------------------------------------------------------------------------
extern "C" void kernel_launch(void* const* d_in, const int* in_sizes, int n_in,
                              void* d_out, int out_size, void* d_ws, size_t ws_size,
                              hipStream_t stream) {
    const float* x        = (const float*)d_in[0];
    const float* W_abcde  = (const float*)d_in[1];
    const float* b_abcde  = (const float*)d_in[2];
    const float* W_O      = (const float*)d_in[3];
    const float* b_O      = (const float*)d_in[4];
    float* out = (float*)d_out;

    float* wsA = (float*)d_ws;                                  // 5*24*1024*64
    float* wsM = wsA + (size_t)5 * BH * SEQ * D_HEAD;           // 24*32*32*128
    float* wsZ = wsM + (size_t)BH * NW * 32 * 128;              // 2048*768

    // 1) abcde = x @ W_abcde + b, scattered into (5, BH, t, d)
    dim3 g1((BS * SEQ) / 64, NCOLS / 64);
    gemm_wmma_f32<0><<<g1, 128, 0, stream>>>(x, W_abcde, b_abcde, wsA,
                                             BS * SEQ, NCOLS, D_MODEL);
    // 2) per-window scores -> softmax marginals
    dim3 g2(NW, BH);
    tritt_scores<<<g2, 256, 0, stream>>>(wsA, wsM);
    // 3) z = sj@la_d + sk@la_e
    tritt_z<<<g2, 256, 0, stream>>>(wsA, wsM, wsZ);
    // 4) out = z @ W_O + b_O
    dim3 g3((BS * SEQ) / 64, D_MODEL / 64);
    gemm_wmma_f32<1><<<g3, 128, 0, stream>>>(wsZ, W_O, b_O, out,
                                             BS * SEQ, D_MODEL, D_MODEL);
}